// SparseChebyKANLayer_53807350284398
// MI455X (gfx1250) — compile-verified
//
#include <hip/hip_runtime.h>
#include <hip/hip_bf16.h>

#define B_SZ   64
#define IN_F   8192
#define OUT_F  8192
#define NNZ_E  262144
#define EPS_C  1e-7f

typedef __attribute__((ext_vector_type(2))) float v2f;
typedef __attribute__((ext_vector_type(8))) float v8f;

// ---------------- zero accumulator + histogram bins ----------------
__global__ void k_zero(float* __restrict__ ws_y, int* __restrict__ counts) {
  int i = blockIdx.x * 256 + threadIdx.x;
  if (i < OUT_F * B_SZ) ws_y[i] = 0.0f;
  if (i < IN_F) counts[i] = 0;
}

// ---------------- histogram of edges per column ----------------
__global__ void k_hist(const int* __restrict__ cols, int* __restrict__ counts) {
  int e = blockIdx.x * 256 + threadIdx.x;
  if (e < NNZ_E) atomicAdd(&counts[cols[e]], 1);
}

// ---------------- exclusive scan over 8192 bins (one block) ----------------
__global__ void __launch_bounds__(256) k_scan(const int* __restrict__ counts,
                                              int* __restrict__ offsets,
                                              int* __restrict__ cursor) {
  __shared__ int part[256];
  int t = threadIdx.x;
  int base = t * 32;
  int local[32];
  int s = 0;
#pragma unroll
  for (int i = 0; i < 32; ++i) { local[i] = counts[base + i]; s += local[i]; }
  part[t] = s;
  __syncthreads();
  for (int off = 1; off < 256; off <<= 1) {
    int v = (t >= off) ? part[t - off] : 0;
    __syncthreads();
    part[t] += v;
    __syncthreads();
  }
  int run = (t == 0) ? 0 : part[t - 1];
#pragma unroll
  for (int i = 0; i < 32; ++i) {
    offsets[base + i] = run;
    cursor[base + i] = run;
    run += local[i];
  }
}

// ---------------- scatter edge ids into column bins ----------------
__global__ void k_scatter(const int* __restrict__ cols, int* __restrict__ cursor,
                          int* __restrict__ sorted) {
  int e = blockIdx.x * 256 + threadIdx.x;
  if (e < NNZ_E) {
    int c = cols[e];
    int pos = atomicAdd(&cursor[c], 1);
    sorted[pos] = e;
  }
}

// ---------------- main WMMA kernel: one block per column ----------------
// D(16 edges x 16 batches) = A(16x4 weights) x B(4x16 cheb basis); two K=4
// f32 WMMAs chained via the C operand cover degrees 0..7 exactly.
__global__ void __launch_bounds__(128) k_spmm(
    const float* __restrict__ x, const float* __restrict__ weight,
    const int* __restrict__ rows, const int* __restrict__ offsets,
    const int* __restrict__ counts, const int* __restrict__ sorted,
    float* __restrict__ ws_y) {
  __shared__ __align__(16) int lds_rows[64];   // 16 row ids per wave

  const int c    = blockIdx.x;
  const int lane = threadIdx.x & 31;
  const int wv   = threadIdx.x >> 5;   // batch block 0..3 (one per wave)
  const int half = lane >> 4;          // lane half selects K pair {0,1} vs {2,3}
  const int n    = lane & 15;          // N (batch within block) / M (edge slot)
  const int bb   = wv * 16 + n;        // absolute batch 0..63
  const int dbase = half * 2;

  // Chebyshev basis T_0..T_7 at t = clip(tanh(x[bb, c])), in registers.
  float t = tanhf(x[bb * IN_F + c]);
  t = fminf(fmaxf(t, -1.0f + EPS_C), 1.0f - EPS_C);
  float T[8];
  T[0] = 1.0f; T[1] = t;
#pragma unroll
  for (int d = 2; d < 8; ++d) T[d] = 2.0f * t * T[d - 1] - T[d - 2];

  // B operand (4x16): VGPR0 = rows {K=0 | K=2} across lane halves, VGPR1 = {K=1 | K=3}.
  v2f blo; blo.x = T[dbase];     blo.y = T[dbase + 1];
  v2f bhi; bhi.x = T[dbase + 4]; bhi.y = T[dbase + 5];

  const int start = offsets[c];
  const int cnt   = counts[c];

  // Software-pipelined edge-id fetch (one chunk ahead).
  int  nextEid   = 0;
  bool nextValid = n < cnt;
  if (nextValid) nextEid = sorted[start + n];

  for (int s = 0; s < cnt; s += 16) {
    const int  eid   = nextEid;
    const bool valid = nextValid;

    v2f alo = {0.0f, 0.0f};
    v2f ahi = {0.0f, 0.0f};
    int row = -1;
    if (valid) {
      const float* wp = weight + eid * 8;
      // A layout: lanes 0-15 -> K=0,1 ; lanes 16-31 -> K=2,3 ; M = lane&15
      alo = *(const v2f*)(wp + dbase);
      ahi = *(const v2f*)(wp + dbase + 4);
      row = rows[eid];
    }

    // Issue next chunk's edge-id load early; prefetch two chunks ahead.
    nextValid = (s + 16 + n) < cnt;
    if (nextValid) nextEid = sorted[start + s + 16 + n];
    __builtin_prefetch(sorted + start + s + 32 + n, 0, 3);

    // Stage this chunk's 16 row ids in the wave's LDS slot (in-order per wave).
    __builtin_amdgcn_wave_barrier();
    if (lane < 16) lds_rows[wv * 16 + n] = row;
    __builtin_amdgcn_wave_barrier();
    const int4* rp = (const int4*)&lds_rows[wv * 16 + half * 8];
    int4 r0 = rp[0];
    int4 r1 = rp[1];
    __builtin_amdgcn_wave_barrier();

    v8f acc = {0.0f, 0.0f, 0.0f, 0.0f, 0.0f, 0.0f, 0.0f, 0.0f};
    acc = __builtin_amdgcn_wmma_f32_16x16x4_f32(false, alo, false, blo,
                                                (short)0, acc, false, false);
    acc = __builtin_amdgcn_wmma_f32_16x16x4_f32(false, ahi, false, bhi,
                                                (short)0, acc, false, false);

    // D: VGPR v -> edge (half ? v+8 : v), batch = lane&15.
    const int rows8[8] = {r0.x, r0.y, r0.z, r0.w, r1.x, r1.y, r1.z, r1.w};
#pragma unroll
    for (int v = 0; v < 8; ++v) {
      int rv = rows8[v];
      float val = acc[v];
      if (rv >= 0) unsafeAtomicAdd(&ws_y[rv * B_SZ + bb], val);
    }
  }
}

// ---------------- transpose accumulator [r][b] -> output [b][r] ----------------
__global__ void k_out(const float* __restrict__ ws_y, float* __restrict__ out) {
  int i = blockIdx.x * 256 + threadIdx.x;   // i = b*OUT_F + r
  int b = i >> 13;
  int r = i & (OUT_F - 1);
  out[i] = ws_y[r * B_SZ + b];
}

extern "C" void kernel_launch(void* const* d_in, const int* in_sizes, int n_in,
                              void* d_out, int out_size, void* d_ws, size_t ws_size,
                              hipStream_t stream) {
  const float* x      = (const float*)d_in[0];
  const float* weight = (const float*)d_in[1];
  const int*   rows   = (const int*)d_in[2];
  const int*   cols   = (const int*)d_in[3];
  float*       out    = (float*)d_out;

  char* ws = (char*)d_ws;
  float* ws_y    = (float*)ws;                                 // 2 MB accumulator [r][b]
  int*   counts  = (int*)(ws + (size_t)OUT_F * B_SZ * sizeof(float));
  int*   offsets = counts + IN_F;
  int*   cursor  = offsets + IN_F;
  int*   sorted  = cursor + IN_F;                              // 1 MB edge ids

  k_zero   <<<(OUT_F * B_SZ) / 256, 256, 0, stream>>>(ws_y, counts);
  k_hist   <<<NNZ_E / 256, 256, 0, stream>>>(cols, counts);
  k_scan   <<<1, 256, 0, stream>>>(counts, offsets, cursor);
  k_scatter<<<NNZ_E / 256, 256, 0, stream>>>(cols, cursor, sorted);
  k_spmm   <<<IN_F, 128, 0, stream>>>(x, weight, rows, offsets, counts, sorted, ws_y);
  k_out    <<<(B_SZ * OUT_F) / 256, 256, 0, stream>>>(ws_y, out);
}